// Erosion2d_14319420965424
// MI455X (gfx1250) — compile-verified
//
#include <hip/hip_runtime.h>
#include <stdint.h>

// 3x3 erosion (sliding-window min) over [B,C,1024,1024] f32, pad value 1e9.
// Memory-bound: ~268 MB HBM traffic -> ~11.5 us floor at 23.3 TB/s.
// CDNA5 path: async global->LDS staging (ASYNCcnt) + LDS-tiled separable min.

#define IMG        1024
#define TILE       64
#define HALO_H     66          // rows gy0-1 .. gy0+64
#define HALO_W     72          // cols gx0-4 .. gx0+67 (16B-aligned vec4 region)
#define LDS_STRIDE 72          // floats per LDS row (mod 64 = 8 -> conflict-free b32 reads)
#define VEC_COLS   (HALO_W / 4)          // 18 vec4 per row
#define NCELLS     (HALO_H * VEC_COLS)   // 1188 vec4 transfers per tile
#define PAD_VAL    1.0e9f

__global__ __launch_bounds__(256) void erosion3x3_kernel(const float* __restrict__ x,
                                                         float* __restrict__ out) {
    __shared__ __align__(16) float L[HALO_H * LDS_STRIDE];

    const int t   = threadIdx.x;
    const int gx0 = blockIdx.x * TILE;
    const int gy0 = blockIdx.y * TILE;
    const size_t plane = (size_t)blockIdx.z * (size_t)(IMG * IMG);
    const float* __restrict__ src = x + plane;

    // ---- Phase 1: stage (66 x 72) halo tile into LDS -----------------------
    // Each vec4 transfer is 16B aligned and fully inside or fully outside the
    // image row (gx0 % 64 == 0, widths multiples of 4). In-image cells use
    // async DMA global->LDS; out-of-image cells get the erosion pad value.
    for (int c = t; c < NCELLS; c += 256) {
        const int cy = c / VEC_COLS;
        const int c4 = c - cy * VEC_COLS;
        const int gy = gy0 - 1 + cy;
        const int gx = gx0 - 4 + c4 * 4;
        float* lcell = &L[cy * LDS_STRIDE + c4 * 4];
        if ((unsigned)gy < (unsigned)IMG && (unsigned)gx < (unsigned)IMG) {
            // LDS byte address = low 32 bits of the flat shared address
            // (allocation-relative offset, what the async-to-LDS VDST expects).
            const uint32_t lds_off = (uint32_t)(uintptr_t)lcell;
            const unsigned long long gaddr =
                (unsigned long long)(uintptr_t)(src + (size_t)gy * IMG + gx);
            asm volatile("global_load_async_to_lds_b128 %0, %1, off"
                         :
                         : "v"(lds_off), "v"(gaddr)
                         : "memory");
        } else {
            // Outside the image -> pad so it never wins the min.
            *reinterpret_cast<float4*>(lcell) =
                make_float4(PAD_VAL, PAD_VAL, PAD_VAL, PAD_VAL);
        }
    }

    // Drain this wave's async transfers, then workgroup barrier (ds stores are
    // compiler-tracked; the asynccnt wait covers the DMA writes to LDS).
    asm volatile("s_wait_asynccnt 0" ::: "memory");
    __syncthreads();

    // ---- Phase 2: separable 3x3 min with vertical register reuse -----------
    // Thread owns output column tx of rows [r0, r0+16) of the tile.
    const int tx = t & 63;
    const int ty = t >> 6;
    const int r0 = ty * 16;

    float* __restrict__ dst =
        out + plane + (size_t)(gy0 + r0) * IMG + (size_t)(gx0 + tx);

    // Horizontal min over window columns: output x=tx uses gx in
    // [gx0+tx-1, gx0+tx+1] -> LDS columns cx = tx+3 .. tx+5.
    #define HMIN(cy_)                                                          \
        ({ const int _b = (cy_) * LDS_STRIDE + tx + 3;                         \
           fminf(fminf(L[_b], L[_b + 1]), L[_b + 2]); })

    float hm0 = HMIN(r0 + 0);
    float hm1 = HMIN(r0 + 1);
    #pragma unroll
    for (int i = 0; i < 16; ++i) {
        const float hm2 = HMIN(r0 + i + 2);
        const float o   = fminf(fminf(hm0, hm1), hm2);
        // Write-once output: non-temporal so the 192MB L2 keeps input halos.
        __builtin_nontemporal_store(o, dst + (size_t)i * IMG);
        hm0 = hm1;
        hm1 = hm2;
    }
    #undef HMIN
}

extern "C" void kernel_launch(void* const* d_in, const int* in_sizes, int n_in,
                              void* d_out, int out_size, void* d_ws, size_t ws_size,
                              hipStream_t stream) {
    const float* x = (const float*)d_in[0];
    float* out = (float*)d_out;

    const int planes = in_sizes[0] / (IMG * IMG);   // 8*4 = 32
    dim3 grid(IMG / TILE, IMG / TILE, planes);      // (16, 16, 32)
    erosion3x3_kernel<<<grid, 256, 0, stream>>>(x, out);
}